// MultiViewMultiPersonPoseNet_12051678232599
// MI455X (gfx1250) — compile-verified
//
#include <hip/hip_runtime.h>
#include <math.h>

// Problem constants (match reference)
#define BB   16
#define NP   64
#define NJ   15
#define ND   128
#define KP   48          // padded K = 3*NJ -> 48
#define BIGV 100000.0f
#define EPSV 1e-8f

typedef float v2f __attribute__((ext_vector_type(2)));
typedef float v8f __attribute__((ext_vector_type(8)));

// ---------------------------------------------------------------------------
// Kernel 1: projection + A operand  A[b][d][p][k], k=3j+{0,1,2} = (pt2,u,v)
// ---------------------------------------------------------------------------
__global__ void build_a(const float* __restrict__ poses3d,
                        const float* __restrict__ R,
                        const float* __restrict__ T,
                        const float* __restrict__ f,
                        const float* __restrict__ c,
                        float* __restrict__ A) {
    int idx = blockIdx.x * blockDim.x + threadIdx.x;   // over B*NP*ND
    if (idx >= BB * NP * ND) return;
    int d = idx % ND;
    int t = idx / ND;
    int p = t % NP;
    int b = t / NP;

    float r00 = R[b*9+0], r01 = R[b*9+1], r02 = R[b*9+2];
    float r10 = R[b*9+3], r11 = R[b*9+4], r12 = R[b*9+5];
    float r20 = R[b*9+6], r21 = R[b*9+7], r22 = R[b*9+8];
    float tx = T[b*3+0], ty = T[b*3+1], tz = T[b*3+2];
    float fx = f[b*2+0], fy = f[b*2+1];
    float cx = c[b*2+0], cy = c[b*2+1];

    long abase = ((long)(b * ND + d) * NP + p) * KP;
    for (int j = 0; j < NJ; ++j) {
        long pi = ((((long)b * NP + p) * NJ + j) * ND + d) * 3;
        float X = poses3d[pi + 0] - tx;
        float Y = poses3d[pi + 1] - ty;
        float Z = poses3d[pi + 2] - tz;
        float xc0 = r00 * X + r01 * Y + r02 * Z;
        float xc1 = r10 * X + r11 * Y + r12 * Z;
        float xc2 = r20 * X + r21 * Y + r22 * Z;
        float u = fx * xc0 / xc2 + cx;
        float v = fy * xc1 / xc2 + cy;
        A[abase + 3*j + 0] = u * u + v * v;
        A[abase + 3*j + 1] = u;
        A[abase + 3*j + 2] = v;
    }
    A[abase + 45] = 0.0f;
    A[abase + 46] = 0.0f;
    A[abase + 47] = 0.0f;
}

// ---------------------------------------------------------------------------
// Kernel 2: B operand Bm[b][q][k] (scaled by 1/(vsum+eps)), plus term2'
// ---------------------------------------------------------------------------
__global__ void build_b(const float* __restrict__ p2ref,
                        const float* __restrict__ vis,
                        float* __restrict__ Bm,
                        float* __restrict__ t2) {
    int idx = blockIdx.x * blockDim.x + threadIdx.x;   // over B*NP
    if (idx >= BB * NP) return;
    int q = idx % NP;
    int b = idx / NP;

    float vsum = 0.0f;
    for (int j = 0; j < NJ; ++j) vsum += vis[(b * NP + q) * NJ + j];
    float inv = 1.0f / (vsum + EPSV);

    long bb = (long)(b * NP + q) * KP;
    float t2v = 0.0f;
    for (int j = 0; j < NJ; ++j) {
        int ri = (b * NP + q) * NJ + j;
        float vv = vis[ri];
        float rx = p2ref[ri * 2 + 0];
        float ry = p2ref[ri * 2 + 1];
        Bm[bb + 3*j + 0] = vv * inv;
        Bm[bb + 3*j + 1] = -2.0f * rx * vv * inv;
        Bm[bb + 3*j + 2] = -2.0f * ry * vv * inv;
        t2v += (rx * rx + ry * ry) * vv;
    }
    Bm[bb + 45] = 0.0f;
    Bm[bb + 46] = 0.0f;
    Bm[bb + 47] = 0.0f;
    t2[idx] = t2v * inv;
}

// ---------------------------------------------------------------------------
// Kernel 3: fused  dist = A x B + term2'  (f32 WMMA 16x16x4), mask, argmin(q),
// gather matched ref pose, score.  One block = (b, pair of d). 8 wave32s:
// wave -> (p_tile = wave&3, d_off = wave>>2).
// ---------------------------------------------------------------------------
__global__ void __launch_bounds__(256)
dist_argmin_score(const float* __restrict__ A,
                  const float* __restrict__ Bm,
                  const float* __restrict__ t2,
                  const int*   __restrict__ npers,
                  const float* __restrict__ p2ref,
                  const float* __restrict__ vis,
                  const float* __restrict__ imw,
                  const float* __restrict__ imh,
                  float* __restrict__ out) {
    __shared__ int mmLds[2 * NP];

    int tid   = threadIdx.x;
    int lane  = tid & 31;
    int wave  = tid >> 5;
    int ptile = wave & 3;
    int doff  = wave >> 2;
    int b     = blockIdx.x >> 6;      // 64 d-pairs per batch
    int dpair = blockIdx.x & 63;
    int d     = dpair * 2 + doff;
    int half  = lane >> 4;            // K sub-split per 32-bit WMMA layouts
    int ln    = lane & 15;

    // A fragment rows: lanes 0-15 / 16-31 -> same M, K offset 0 / 2
    const float* Arow = A + ((long)(b * ND + d) * NP + ptile * 16 + ln) * KP + half * 2;
    // B fragment cols: lane ln -> column q = qtile*16+ln, K offset half*2
    const float* Brow = Bm + (long)b * NP * KP + (long)ln * KP + half * 2;

    v8f acc0 = {}, acc1 = {}, acc2 = {}, acc3 = {};
    #pragma unroll
    for (int k0 = 0; k0 < KP; k0 += 4) {
        v2f a  = *(const v2f*)(Arow + k0);
        v2f b0 = *(const v2f*)(Brow + 0 * 16 * KP + k0);
        v2f b1 = *(const v2f*)(Brow + 1 * 16 * KP + k0);
        v2f b2 = *(const v2f*)(Brow + 2 * 16 * KP + k0);
        v2f b3 = *(const v2f*)(Brow + 3 * 16 * KP + k0);
        acc0 = __builtin_amdgcn_wmma_f32_16x16x4_f32(false, a, false, b0, (short)0, acc0, false, false);
        acc1 = __builtin_amdgcn_wmma_f32_16x16x4_f32(false, a, false, b1, (short)0, acc1, false, false);
        acc2 = __builtin_amdgcn_wmma_f32_16x16x4_f32(false, a, false, b2, (short)0, acc2, false, false);
        acc3 = __builtin_amdgcn_wmma_f32_16x16x4_f32(false, a, false, b3, (short)0, acc3, false, false);
    }

    // Per-q additive term and validity mask
    int np = npers[b];
    float t2v[4];
    bool  ok[4];
    #pragma unroll
    for (int t = 0; t < 4; ++t) {
        int q = t * 16 + ln;
        t2v[t] = t2[b * NP + q];
        ok[t]  = q < np;
    }

    // Per-lane argmin over the 4 q-tiles, then butterfly within each 16-lane
    // half (half 0 holds rows M=r, half 1 holds rows M=8+r).
    float bestv[8];
    int   bestq[8];
    #pragma unroll
    for (int r = 0; r < 8; ++r) {
        float v0 = ok[0] ? (acc0[r] + t2v[0]) : BIGV;
        float v1 = ok[1] ? (acc1[r] + t2v[1]) : BIGV;
        float v2 = ok[2] ? (acc2[r] + t2v[2]) : BIGV;
        float v3 = ok[3] ? (acc3[r] + t2v[3]) : BIGV;
        float bv = v0; int bq = 0 * 16 + ln;
        if (v1 < bv) { bv = v1; bq = 1 * 16 + ln; }
        if (v2 < bv) { bv = v2; bq = 2 * 16 + ln; }
        if (v3 < bv) { bv = v3; bq = 3 * 16 + ln; }
        bestv[r] = bv; bestq[r] = bq;
    }
    #pragma unroll
    for (int off = 1; off < 16; off <<= 1) {
        #pragma unroll
        for (int r = 0; r < 8; ++r) {
            float ov = __shfl_xor(bestv[r], off, 32);
            int   oq = __shfl_xor(bestq[r], off, 32);
            if (ov < bestv[r] || (ov == bestv[r] && oq < bestq[r])) {
                bestv[r] = ov; bestq[r] = oq;
            }
        }
    }
    if (ln == 0) {
        #pragma unroll
        for (int r = 0; r < 8; ++r)
            mmLds[doff * NP + ptile * 16 + half * 8 + r] = bestq[r];
    }
    __syncthreads();

    // Final gather + score for this block's 2 d-values x 64 p x 15 j
    float iw = imw[b] - 1.0f;
    float ih = imh[b] - 1.0f;
    for (int i = tid; i < 2 * NP * NJ; i += 256) {
        int dd  = i / (NP * NJ);
        int rem = i - dd * (NP * NJ);
        int p   = rem / NJ;
        int j   = rem - p * NJ;
        int d2  = dpair * 2 + dd;
        int q   = mmLds[dd * NP + p];

        const float* ar = A + ((long)(b * ND + d2) * NP + p) * KP + 3 * j;
        float u = ar[1];
        float v = ar[2];

        int ri = (b * NP + q) * NJ + j;
        float rx = p2ref[ri * 2 + 0];
        float ry = p2ref[ri * 2 + 1];
        float vv = vis[ri];

        float du = u - rx, dv = v - ry;
        float s = expf(-(du * du + dv * dv) * (1.0f / 225.0f));
        bool inb = (u >= 0.0f) && (v >= 0.0f) && (u <= iw) && (v <= ih);
        out[(((long)b * NP + p) * NJ + j) * ND + d2] = inb ? s * vv : 0.0f;
    }
}

// ---------------------------------------------------------------------------
extern "C" void kernel_launch(void* const* d_in, const int* in_sizes, int n_in,
                              void* d_out, int out_size, void* d_ws, size_t ws_size,
                              hipStream_t stream) {
    const float* poses3d = (const float*)d_in[0];
    const float* p2ref   = (const float*)d_in[1];
    const float* visr    = (const float*)d_in[2];
    const int*   npers   = (const int*)  d_in[3];
    const float* R       = (const float*)d_in[4];
    const float* T       = (const float*)d_in[5];
    const float* f       = (const float*)d_in[6];
    const float* c       = (const float*)d_in[7];
    const float* imw     = (const float*)d_in[8];
    const float* imh     = (const float*)d_in[9];
    float* out = (float*)d_out;

    // Workspace partition
    float* A  = (float*)d_ws;                         // BB*ND*NP*KP
    float* Bm = A + (size_t)BB * ND * NP * KP;        // BB*NP*KP
    float* t2 = Bm + (size_t)BB * NP * KP;            // BB*NP

    int nA = BB * NP * ND;                            // 131072
    build_a<<<(nA + 255) / 256, 256, 0, stream>>>(poses3d, R, T, f, c, A);

    int nB = BB * NP;                                 // 1024
    build_b<<<(nB + 255) / 256, 256, 0, stream>>>(p2ref, visr, Bm, t2);

    int nBlocks = BB * (ND / 2);                      // 1024 blocks, 8 waves each
    dist_argmin_score<<<nBlocks, 256, 0, stream>>>(A, Bm, t2, npers, p2ref, visr,
                                                   imw, imh, out);
}